// Network_53807350284628
// MI455X (gfx1250) — compile-verified
//
// ============================================================================
// DARTS supernet forward — AMD MI455X (gfx1250, CDNA5), wave32 / WMMA.
//
// Design (compile-only reasoning from MI455X specs):
//  * Whole net fits in L2 (192MB); latency-bound, not HBM-bound.
//  * All 1x1 convs (the FLOP carriers) -> V_WMMA_F32_16X16X4_F32, one wave32
//    per 16(out-ch) x 16(pixel) tile, K stepped by 4 input channels.
//    K loop is branch-free (Cin % 4 == 0 at every call site) and unrolled x2.
//  * BN stats reductions stage data via global_load_async_to_lds_b128 +
//    s_wait_asynccnt (CDNA5 async path).
//  * Depthwise/pool/gate kernels are VALU grid-stride (negligible cost).
//  * Host walks the params flat buffer in jax tree-leaves order (sorted keys).
// ============================================================================
#include <hip/hip_runtime.h>
#include <hip/hip_fp16.h>
#include <stdint.h>

typedef __attribute__((ext_vector_type(2))) float v2f;
typedef __attribute__((ext_vector_type(8))) float v8f;

#define GS(i, n) for (size_t i = (size_t)blockIdx.x * blockDim.x + threadIdx.x; \
                      i < (n); i += (size_t)gridDim.x * blockDim.x)

static inline int gblk(size_t n) { size_t b = (n + 255) / 256; return (int)(b > 1048576 ? 1048576 : b); }

// ---------------------------------------------------------------------------
// Elementwise utility kernels
// ---------------------------------------------------------------------------
__global__ void k_zero(float* p, size_t n) { GS(i, n) p[i] = 0.f; }
__global__ void k_copy(float* d, const float* s, size_t n) { GS(i, n) d[i] = s[i]; }
// d += w[wi] * s   (w is a device-resident softmax vector)
__global__ void k_axpy(float* d, const float* s, size_t n, const float* w, int wi) {
  float a = w[wi];
  GS(i, n) d[i] += a * s[i];
}
// xg[n,c,hw] = x[n,c,hw] * gate[n,c]
__global__ void k_apply_gate(const float* x, const float* gate, float* xg,
                             int Nb, int C, int HW) {
  size_t tot = (size_t)Nb * C * HW;
  GS(i, tot) {
    int c = (int)((i / HW) % C);
    int n = (int)(i / ((size_t)HW * C));
    xg[i] = x[i] * gate[n * C + c];
  }
}
// xt[n,j,hw] = xg[n,idx[j],hw]
__global__ void k_gather(const float* xg, float* xt, const int* idx,
                         int Nb, int C, int Cop, int HW) {
  size_t tot = (size_t)Nb * Cop * HW;
  GS(i, tot) {
    int hw = (int)(i % HW);
    int j = (int)((i / HW) % Cop);
    int n = (int)(i / ((size_t)HW * Cop));
    xt[i] = xg[((size_t)n * C + idx[j]) * HW + hw];
  }
}
// dst[n,idx[j],hw] = src[n,j,hw]
__global__ void k_scatter(float* dst, const float* src, const int* idx,
                          int Nb, int C, int Cop, int HW) {
  size_t tot = (size_t)Nb * Cop * HW;
  GS(i, tot) {
    int hw = (int)(i % HW);
    int j = (int)((i / HW) % Cop);
    int n = (int)(i / ((size_t)HW * Cop));
    dst[((size_t)n * C + idx[j]) * HW + hw] = src[i];
  }
}
// channel-slice copy for cell-output concat
__global__ void k_copy_ch(float* dst, const float* src, int coBase,
                          int Csrc, int Cdst, int HW, int Nb) {
  size_t tot = (size_t)Nb * Csrc * HW;
  GS(i, tot) {
    int hw = (int)(i % HW);
    int c = (int)((i / HW) % Csrc);
    int n = (int)(i / ((size_t)HW * Csrc));
    dst[((size_t)n * Cdst + coBase + c) * HW + hw] = src[i];
  }
}

// ---------------------------------------------------------------------------
// WMMA fp32 pointwise (1x1) conv:  Y[co,p] = sum_ci W[co,ci] * act(X[ci,p])
// One wave32 computes a 16(out-ch) x 16(pixel) tile with V_WMMA_F32_16X16X4_F32.
// A (16x4 f32) layout: lanes 0-15 hold {K0,K1}, lanes 16-31 hold {K2,K3}.
// D (16x16 f32) layout: VGPR r -> row r (lanes 0-15) / row r+8 (lanes 16-31).
// stride/off implement the factorized-reduce paths (stride-2, (0,0)/(1,1)).
// PRECONDITION: Cin % 4 == 0 (true at every call site) -> branch-free K loop.
// ---------------------------------------------------------------------------
__global__ void k_pw_wmma(const float* __restrict__ X, const float* __restrict__ Wt,
                          float* __restrict__ Y, int Nb, int Cin, int Cout,
                          int Hin, int Win, int stride, int off,
                          int coBase, int CoutTot, int doRelu) {
  int Ho = Hin / stride, Wo = Win / stride;
  int HWo = Ho * Wo, HWi = Hin * Win;
  int tilesPerImg = HWo >> 4;                  // HW is always a multiple of 16
  int tile = blockIdx.x % tilesPerImg;
  int n = blockIdx.x / tilesPerImg;
  int coT = blockIdx.y << 4;
  int lane = threadIdx.x & 31;
  int half = lane >> 4;                        // 0: A K0/K1, B K0/K2; 1: A K2/K3, B K1/K3
  int l = lane & 15;
  int s0 = tile << 4;                          // output-pixel base within image
  int so = s0 + l;
  int hi = (so / Wo) * stride + off;
  int wi = (so % Wo) * stride + off;
  int pin = hi * Win + wi;                     // == so when stride==1 && off==0
  // A-row: clamp+select (branch-free) for partial tiles (Cout < 16).
  int row = coT + l;
  bool rowOK = row < Cout;
  int rowc = rowOK ? row : (Cout - 1);
  const float* Wrow = Wt + (size_t)rowc * Cin + half * 2; // 8B-aligned pairs
  const float* Bp = X + (size_t)n * Cin * HWi + (size_t)half * HWi + pin;
  __builtin_prefetch(Wt + (size_t)coT * Cin, 0, 1);   // -> global_prefetch_b8
  v8f acc = {};
#pragma unroll 2
  for (int k = 0; k < Cin; k += 4) {
    v2f wv = *(const v2f*)(Wrow + k);          // global_load_b64
    v2f a, b;
    a.x = rowOK ? wv.x : 0.f;                  // v_cndmask, no branches
    a.y = rowOK ? wv.y : 0.f;
    float b0 = Bp[(size_t)k * HWi];            // unconditional coalesced loads
    float b1 = Bp[(size_t)(k + 2) * HWi];
    if (doRelu) { b0 = fmaxf(b0, 0.f); b1 = fmaxf(b1, 0.f); }
    b.x = b0; b.y = b1;
    acc = __builtin_amdgcn_wmma_f32_16x16x4_f32(false, a, false, b,
                                                (short)0, acc, false, false);
  }
  float* Yn = Y + (size_t)n * CoutTot * HWo;
#pragma unroll
  for (int r = 0; r < 8; ++r) {
    int co = coT + r + half * 8;
    if (co < Cout) Yn[(size_t)(coBase + co) * HWo + s0 + l] = acc[r];
  }
}

// ---------------------------------------------------------------------------
// General direct conv (stem, depthwise, dilated-depthwise, 1x7 / 7x1).
// Optional ReLU applied to the *input* (matches relu-before-conv in reference).
// ---------------------------------------------------------------------------
__global__ void k_conv(const float* __restrict__ X, const float* __restrict__ Wt,
                       float* __restrict__ Y, int Nb, int Cin, int Cout,
                       int H, int W, int KH, int KW, int SH, int SW,
                       int PH, int PW, int DH, int DWd, int groups, int doRelu,
                       int coBase, int CoutTot) {
  int Ho = (H + 2 * PH - DH * (KH - 1) - 1) / SH + 1;
  int Wo = (W + 2 * PW - DWd * (KW - 1) - 1) / SW + 1;
  int cig = Cin / groups, cog = Cout / groups;
  size_t tot = (size_t)Nb * Cout * Ho * Wo;
  GS(t, tot) {
    int wo = (int)(t % Wo);
    int ho = (int)((t / Wo) % Ho);
    int co = (int)((t / ((size_t)Wo * Ho)) % Cout);
    int n = (int)(t / ((size_t)Wo * Ho * Cout));
    int g = co / cog;
    float acc = 0.f;
    for (int ci = 0; ci < cig; ++ci) {
      int c = g * cig + ci;
      for (int kh = 0; kh < KH; ++kh) {
        int hi = ho * SH - PH + kh * DH;
        if (hi < 0 || hi >= H) continue;
        for (int kw = 0; kw < KW; ++kw) {
          int wi = wo * SW - PW + kw * DWd;
          if (wi < 0 || wi >= W) continue;
          float v = X[(((size_t)n * Cin + c) * H + hi) * W + wi];
          if (doRelu) v = fmaxf(v, 0.f);
          acc += Wt[(((size_t)co * cig + ci) * KH + kh) * KW + kw] * v;
        }
      }
    }
    Y[(((size_t)n * CoutTot + coBase + co) * Ho + ho) * Wo + wo] = acc;
  }
}

// ---------------------------------------------------------------------------
// Pooling (max 3x3/2x2, avg 3x3). avg divides by k*k with zero padding (ref).
// ---------------------------------------------------------------------------
__global__ void k_pool(const float* X, float* Y, int Nb, int C, int H, int W,
                       int k, int s, int p, int isMax) {
  int Ho = (H + 2 * p - k) / s + 1, Wo = (W + 2 * p - k) / s + 1;
  size_t tot = (size_t)Nb * C * Ho * Wo;
  float inv = 1.f / (float)(k * k);
  GS(t, tot) {
    int wo = (int)(t % Wo);
    int ho = (int)((t / Wo) % Ho);
    int c = (int)((t / ((size_t)Wo * Ho)) % C);
    int n = (int)(t / ((size_t)Wo * Ho * C));
    float acc = isMax ? -1e30f : 0.f;
    for (int kh = 0; kh < k; ++kh) {
      int hi = ho * s - p + kh;
      if (hi < 0 || hi >= H) continue;
      for (int kw = 0; kw < k; ++kw) {
        int wi = wo * s - p + kw;
        if (wi < 0 || wi >= W) continue;
        float v = X[(((size_t)n * C + c) * H + hi) * W + wi];
        acc = isMax ? fmaxf(acc, v) : acc + v;
      }
    }
    Y[t] = isMax ? acc : acc * inv;
  }
}

// ---------------------------------------------------------------------------
// BN statistics: per-channel mean / rstd over (N, H, W).  Global data is
// staged into LDS with CDNA5 async loads (global_load_async_to_lds_b128),
// then reduced; one workgroup per channel.
// ---------------------------------------------------------------------------
__global__ void k_bn_stats(const float* __restrict__ X, float* mean, float* rstd,
                           int Nb, int C, int HW, float eps) {
  __shared__ float sbuf[1024];
  __shared__ float r1[256], r2[256];
  int c = blockIdx.x;
  int tid = threadIdx.x;
  float s = 0.f, ss = 0.f;
  for (int n = 0; n < Nb; ++n) {
    const float* base = X + ((size_t)n * C + c) * HW;
    for (int i0 = 0; i0 < HW; i0 += 1024) {
      int idx = i0 + tid * 4;
      if (idx < HW) {
        uint32_t ldsOff = (uint32_t)(size_t)(&sbuf[tid * 4]);
        const float* g = base + idx;
        asm volatile("global_load_async_to_lds_b128 %0, %1, off"
                     :: "v"(ldsOff), "v"(g) : "memory");
      }
      asm volatile("s_wait_asynccnt 0" ::: "memory");
      __syncthreads();
      int lim = HW - i0; if (lim > 1024) lim = 1024;
      for (int j = tid; j < lim; j += 256) { float v = sbuf[j]; s += v; ss += v * v; }
      __syncthreads();
    }
  }
  r1[tid] = s; r2[tid] = ss; __syncthreads();
  for (int st = 128; st > 0; st >>= 1) {
    if (tid < st) { r1[tid] += r1[tid + st]; r2[tid] += r2[tid + st]; }
    __syncthreads();
  }
  if (tid == 0) {
    float cnt = (float)Nb * (float)HW;
    float m = r1[0] / cnt;
    float v = r2[0] / cnt - m * m;
    mean[c] = m; rstd[c] = rsqrtf(v + eps);
  }
}
__global__ void k_bn_apply(float* X, const float* mean, const float* rstd,
                           const float* g, const float* b, int Nb, int C, int HW) {
  size_t tot = (size_t)Nb * C * HW;
  GS(i, tot) {
    int c = (int)((i / HW) % C);
    float y = (X[i] - mean[c]) * rstd[c];
    if (g) y = y * g[c] + b[c];
    X[i] = y;
  }
}

// ---------------------------------------------------------------------------
// Channel-attention gating pipeline (tiny matmuls, one block per batch row)
// ---------------------------------------------------------------------------
__global__ void k_featpool(const float* X, float* feat, int Nb, int C, int HW) {
  int bc = blockIdx.x, b = bc / C, c = bc % C, tid = threadIdx.x;   // 64 threads
  const float* p = X + ((size_t)b * C + c) * HW;
  float s = 0.f, m = -1e30f;
  for (int i = tid; i < HW; i += 64) { float v = p[i]; s += v; m = fmaxf(m, v); }
  __shared__ float rs[64], rm[64];
  rs[tid] = s; rm[tid] = m; __syncthreads();
  for (int st = 32; st > 0; st >>= 1) {
    if (tid < st) { rs[tid] += rs[tid + st]; rm[tid] = fmaxf(rm[tid], rm[tid + st]); }
    __syncthreads();
  }
  if (tid == 0) { feat[b * 2 * C + c] = rs[0] / (float)HW; feat[b * 2 * C + C + c] = rm[0]; }
}
__global__ void k_gate(const float* feat, const float* W1, const float* G,
                       const float* W2, float* gate, int Nb, int C) {
  int b = blockIdx.x, tid = threadIdx.x;   // 64 threads
  __shared__ float h[10], t[10];
  if (tid < 10) {
    float s = 0.f;
    for (int f = 0; f < 2 * C; ++f) s += feat[b * 2 * C + f] * W1[tid * 2 * C + f];
    h[tid] = s;
  }
  __syncthreads();
  if (tid < 10) {
    float s = 0.f;
    for (int j = 0; j < 10; ++j) s += h[j] * G[tid * 10 + j];
    t[tid] = fmaxf(s, 0.f);
  }
  __syncthreads();
  for (int c = tid; c < C; c += 64) {
    float s = 0.f;
    for (int i = 0; i < 10; ++i) s += t[i] * W2[c * 10 + i];
    gate[b * C + c] = 1.f / (1.f + __expf(-s));
  }
}
__global__ void k_topk(const float* gate, int* idx, int Nb, int C, int K) {
  __shared__ float sl[64];
  int tid = threadIdx.x;    // 64 threads, 1 block
  if (tid < C) { float s = 0.f; for (int b = 0; b < Nb; ++b) s += gate[b * C + tid]; sl[tid] = s; }
  __syncthreads();
  if (tid == 0) {
    bool used[64]; for (int c = 0; c < C; ++c) used[c] = false;
    for (int k = 0; k < K; ++k) {
      float best = -1e30f; int bi = 0;
      for (int c = 0; c < C; ++c)
        if (!used[c] && sl[c] > best) { best = sl[c]; bi = c; }
      used[bi] = true; idx[k] = bi;
    }
  }
}

// ---------------------------------------------------------------------------
// Alpha / beta preprocessing
// ---------------------------------------------------------------------------
__global__ void k_softmax10(const float* a, float* w) {   // 14 blocks, 1 thread
  int r = blockIdx.x;
  if (threadIdx.x) return;
  float mx = -1e30f;
  for (int i = 0; i < 10; ++i) mx = fmaxf(mx, a[r * 10 + i]);
  float s = 0.f, e[10];
  for (int i = 0; i < 10; ++i) { e[i] = __expf(a[r * 10 + i] - mx); s += e[i]; }
  for (int i = 0; i < 10; ++i) w[r * 10 + i] = e[i] / s;
}
__global__ void k_gram(const float* w, float* G) {        // 1 block, 128 threads
  int t = threadIdx.x;
  if (t < 100) {
    int i = t / 10, j = t % 10;
    float s = 0.f;
    for (int e = 0; e < 14; ++e) s += w[e * 10 + i] * w[e * 10 + j];
    G[i * 10 + j] = s;
  }
}
__global__ void k_beta(const float* b, float* w2) {       // 1 thread
  if (threadIdx.x || blockIdx.x) return;
  int start = 0;
  for (int seg = 0; seg < 4; ++seg) {
    int n = 2 + seg;
    float mx = -1e30f;
    for (int i = 0; i < n; ++i) mx = fmaxf(mx, b[start + i]);
    float s = 0.f;
    for (int i = 0; i < n; ++i) s += __expf(b[start + i] - mx);
    for (int i = 0; i < n; ++i) w2[start + i] = __expf(b[start + i] - mx) / s;
    start += n;
  }
}

// ---------------------------------------------------------------------------
// Classifier head: global average pool + linear
// ---------------------------------------------------------------------------
__global__ void k_head(const float* X, const float* Wc, const float* bc,
                       float* out, int Nb, int Cf, int HW) {
  int b = blockIdx.x, tid = threadIdx.x;  // 256 threads
  __shared__ float pooled[256];
  for (int c = tid; c < Cf; c += 256) {
    float s = 0.f;
    const float* p = X + ((size_t)b * Cf + c) * HW;
    for (int i = 0; i < HW; ++i) s += p[i];
    pooled[c] = s / (float)HW;
  }
  __syncthreads();
  if (tid < 10) {
    float s = bc[tid];
    for (int c = 0; c < Cf; ++c) s += pooled[c] * Wc[tid * Cf + c];
    out[b * 10 + tid] = s;
  }
}

// ===========================================================================
// Host side
// ===========================================================================
struct PrimOff { size_t t[4]; };
struct EdgeOff { size_t w1, w2; PrimOff pr[10]; };
struct CellOff { EdgeOff e[14]; size_t p0a, p0b, p1; };

// Replicates setup_inputs() param creation, flattened in jax tree-leaves order
// (sorted dict keys: cells < cls < stem; per cell: ops < p0 < p1; per op:
// ca{w1,w2} < prims[PRIMS order]{sorted keys}).
static size_t walkCell(size_t cur, int C, bool red, bool rp, int cpp, int cprev,
                       CellOff& co) {
  int Cop = C / 4;
  int e = 0;
  for (int i = 0; i < 4; ++i)
    for (int j = 0; j < 2 + i; ++j, ++e) {
      int stride = (red && j < 2) ? 2 : 1;
      EdgeOff& E = co.e[e];
      E.w1 = cur; cur += (size_t)10 * 2 * C;     // ca.w1 (NP, 2C)
      E.w2 = cur; cur += (size_t)C * 10;         // ca.w2 (C, NP)
      for (int p = 0; p < 10; ++p) {
        PrimOff& P = E.pr[p];
        for (int q = 0; q < 4; ++q) P.t[q] = 0;
        if (p == 3) {                             // skip_connect
          if (stride != 1) {
            P.t[0] = cur; cur += (size_t)(Cop / 2) * Cop;   // w1
            P.t[1] = cur; cur += (size_t)(Cop / 2) * Cop;   // w2
          }
        } else if (p >= 4 && p <= 6) {            // sep_conv k: dw1<dw2<pw1<pw2
          int k = 3 + 2 * (p - 4);
          P.t[0] = cur; cur += (size_t)Cop * k * k;
          P.t[1] = cur; cur += (size_t)Cop * k * k;
          P.t[2] = cur; cur += (size_t)Cop * Cop;
          P.t[3] = cur; cur += (size_t)Cop * Cop;
        } else if (p == 7 || p == 8) {            // dil_conv k: dw<pw
          int k = 3 + 2 * (p - 7);
          P.t[0] = cur; cur += (size_t)Cop * k * k;
          P.t[1] = cur; cur += (size_t)Cop * Cop;
        } else if (p == 9) {                      // conv_7x1_1x7: w1<w2
          P.t[0] = cur; cur += (size_t)Cop * Cop * 7;
          P.t[1] = cur; cur += (size_t)Cop * Cop * 7;
        }
      }
    }
  if (rp) {
    co.p0a = cur; cur += (size_t)(C / 2) * cpp;
    co.p0b = cur; cur += (size_t)(C / 2) * cpp;
  } else {
    co.p0a = cur; cur += (size_t)C * cpp; co.p0b = 0;
  }
  co.p1 = cur; cur += (size_t)C * cprev;
  return cur;
}

struct WSRef {
  float *feat, *gate, *xg, *xt, *t1, *t2, *outAcc, *mean, *rstd;
  int* idx;
};

static void pw(const float* X, const float* Wt, float* Y, int Nb, int Cin,
               int Cout, int H, int W, int stride, int off, int coBase,
               int CoutTot, int relu, hipStream_t st) {
  int Ho = H / stride, Wo = W / stride;
  dim3 g((Ho * Wo / 16) * Nb, (Cout + 15) / 16);
  k_pw_wmma<<<g, 32, 0, st>>>(X, Wt, Y, Nb, Cin, Cout, H, W, stride, off,
                              coBase, CoutTot, relu);
}
static void conv(const float* X, const float* Wt, float* Y, int Nb, int Cin,
                 int Cout, int H, int W, int KH, int KW, int SH, int SW,
                 int PH, int PW, int DH, int DWd, int groups, int relu,
                 hipStream_t st) {
  int Ho = (H + 2 * PH - DH * (KH - 1) - 1) / SH + 1;
  int Wo = (W + 2 * PW - DWd * (KW - 1) - 1) / SW + 1;
  size_t tot = (size_t)Nb * Cout * Ho * Wo;
  k_conv<<<gblk(tot), 256, 0, st>>>(X, Wt, Y, Nb, Cin, Cout, H, W, KH, KW,
                                    SH, SW, PH, PW, DH, DWd, groups, relu, 0, Cout);
}
static void bnrun(float* buf, int Nb, int C, int HW, float* mean, float* rstd,
                  const float* g, const float* b, hipStream_t st) {
  k_bn_stats<<<C, 256, 0, st>>>(buf, mean, rstd, Nb, C, HW, 1e-5f);
  size_t tot = (size_t)Nb * C * HW;
  k_bn_apply<<<gblk(tot), 256, 0, st>>>(buf, mean, rstd, g, b, Nb, C, HW);
}

// One mixed-op edge: gating + top-k + 10 primitives + scatter.
static void runMixed(const float* P, const float* x, int Nb, int C, int H, int W,
                     int s, const EdgeOff& E, const float* wv, int ebase,
                     const float* Gm, float* mixedOut, WSRef& ws, hipStream_t st) {
  int Cop = C / 4, HWin = H * W, Ho = H / s, Wo = W / s, HWo = Ho * Wo;
  k_featpool<<<Nb * C, 64, 0, st>>>(x, ws.feat, Nb, C, HWin);
  k_gate<<<Nb, 64, 0, st>>>(ws.feat, P + E.w1, Gm, P + E.w2, ws.gate, Nb, C);
  k_topk<<<1, 64, 0, st>>>(ws.gate, ws.idx, Nb, C, Cop);
  size_t nXG = (size_t)Nb * C * HWin;
  k_apply_gate<<<gblk(nXG), 256, 0, st>>>(x, ws.gate, ws.xg, Nb, C, HWin);
  size_t nXT = (size_t)Nb * Cop * HWin;
  k_gather<<<gblk(nXT), 256, 0, st>>>(ws.xg, ws.xt, ws.idx, Nb, C, Cop, HWin);
  size_t nOut = (size_t)Nb * Cop * HWo;
  k_zero<<<gblk(nOut), 256, 0, st>>>(ws.outAcc, nOut);

  // prim 1: max_pool_3x3 + bn
  k_pool<<<gblk(nOut), 256, 0, st>>>(ws.xt, ws.t1, Nb, Cop, H, W, 3, s, 1, 1);
  bnrun(ws.t1, Nb, Cop, HWo, ws.mean, ws.rstd, nullptr, nullptr, st);
  k_axpy<<<gblk(nOut), 256, 0, st>>>(ws.outAcc, ws.t1, nOut, wv, ebase + 1);
  // prim 2: avg_pool_3x3 + bn
  k_pool<<<gblk(nOut), 256, 0, st>>>(ws.xt, ws.t1, Nb, Cop, H, W, 3, s, 1, 0);
  bnrun(ws.t1, Nb, Cop, HWo, ws.mean, ws.rstd, nullptr, nullptr, st);
  k_axpy<<<gblk(nOut), 256, 0, st>>>(ws.outAcc, ws.t1, nOut, wv, ebase + 2);
  // prim 3: skip_connect
  if (s == 1) {
    k_axpy<<<gblk(nOut), 256, 0, st>>>(ws.outAcc, ws.xt, nOut, wv, ebase + 3);
  } else {
    pw(ws.xt, P + E.pr[3].t[0], ws.t1, Nb, Cop, Cop / 2, H, W, 2, 0, 0, Cop, 1, st);
    pw(ws.xt, P + E.pr[3].t[1], ws.t1, Nb, Cop, Cop / 2, H, W, 2, 1, Cop / 2, Cop, 1, st);
    bnrun(ws.t1, Nb, Cop, HWo, ws.mean, ws.rstd, nullptr, nullptr, st);
    k_axpy<<<gblk(nOut), 256, 0, st>>>(ws.outAcc, ws.t1, nOut, wv, ebase + 3);
  }
  // prims 4..6: sep_conv k = 3/5/7
  for (int p = 4; p <= 6; ++p) {
    int k = 3 + 2 * (p - 4), pd = k / 2;
    conv(ws.xt, P + E.pr[p].t[0], ws.t1, Nb, Cop, Cop, H, W, k, k, s, s, pd, pd, 1, 1, Cop, 1, st);
    pw(ws.t1, P + E.pr[p].t[2], ws.t2, Nb, Cop, Cop, Ho, Wo, 1, 0, 0, Cop, 0, st);
    bnrun(ws.t2, Nb, Cop, HWo, ws.mean, ws.rstd, nullptr, nullptr, st);
    conv(ws.t2, P + E.pr[p].t[1], ws.t1, Nb, Cop, Cop, Ho, Wo, k, k, 1, 1, pd, pd, 1, 1, Cop, 1, st);
    pw(ws.t1, P + E.pr[p].t[3], ws.t2, Nb, Cop, Cop, Ho, Wo, 1, 0, 0, Cop, 0, st);
    bnrun(ws.t2, Nb, Cop, HWo, ws.mean, ws.rstd, nullptr, nullptr, st);
    k_axpy<<<gblk(nOut), 256, 0, st>>>(ws.outAcc, ws.t2, nOut, wv, ebase + p);
  }
  // prims 7,8: dil_conv k = 3/5
  for (int p = 7; p <= 8; ++p) {
    int k = 3 + 2 * (p - 7), pd = (k / 2) * 2;
    conv(ws.xt, P + E.pr[p].t[0], ws.t1, Nb, Cop, Cop, H, W, k, k, s, s, pd, pd, 2, 2, Cop, 1, st);
    pw(ws.t1, P + E.pr[p].t[1], ws.t2, Nb, Cop, Cop, Ho, Wo, 1, 0, 0, Cop, 0, st);
    bnrun(ws.t2, Nb, Cop, HWo, ws.mean, ws.rstd, nullptr, nullptr, st);
    k_axpy<<<gblk(nOut), 256, 0, st>>>(ws.outAcc, ws.t2, nOut, wv, ebase + p);
  }
  // prim 9: conv_7x1_1x7  (1x7 stride (1,s) then 7x1 stride (s,1))
  conv(ws.xt, P + E.pr[9].t[0], ws.t1, Nb, Cop, Cop, H, W, 1, 7, 1, s, 0, 3, 1, 1, 1, 1, st);
  conv(ws.t1, P + E.pr[9].t[1], ws.t2, Nb, Cop, Cop, H, W / s, 7, 1, s, 1, 3, 0, 1, 1, 1, 0, st);
  bnrun(ws.t2, Nb, Cop, HWo, ws.mean, ws.rstd, nullptr, nullptr, st);
  k_axpy<<<gblk(nOut), 256, 0, st>>>(ws.outAcc, ws.t2, nOut, wv, ebase + 9);

  // base = gated x (maxpool-2 when reducing), then scatter top-k channels
  if (s == 1) {
    k_copy<<<gblk(nXG), 256, 0, st>>>(mixedOut, ws.xg, nXG);
  } else {
    size_t nB = (size_t)Nb * C * HWo;
    k_pool<<<gblk(nB), 256, 0, st>>>(ws.xg, mixedOut, Nb, C, H, W, 2, 2, 0, 1);
  }
  k_scatter<<<gblk(nOut), 256, 0, st>>>(mixedOut, ws.outAcc, ws.idx, Nb, C, Cop, HWo);
}

extern "C" void kernel_launch(void* const* d_in, const int* in_sizes, int n_in,
                              void* d_out, int out_size, void* d_ws, size_t ws_size,
                              hipStream_t stream) {
  const float* x  = (const float*)d_in[0];
  const float* an = (const float*)d_in[1];
  const float* ar = (const float*)d_in[2];
  const float* bnv = (const float*)d_in[3];
  const float* brv = (const float*)d_in[4];
  const float* P  = (const float*)d_in[5];
  float* out = (float*)d_out;
  float* W = (float*)d_ws;
  const int Nb = in_sizes[0] / (3 * 32 * 32);

  // ---- workspace carve-out (floats) ----
  size_t o = 0;
  const size_t SLOT_ACT = 4194304, SLOT_PRE = 2097152, SLOT_ST = 1048576, SLOT_T = 524288;
  float* slots[3] = { W + o, W + o + SLOT_ACT, W + o + 2 * SLOT_ACT }; o += 3 * SLOT_ACT;
  float* pre0 = W + o; o += SLOT_PRE;
  float* pre1 = W + o; o += SLOT_PRE;
  float* st4[4]; for (int i = 0; i < 4; ++i) { st4[i] = W + o; o += SLOT_ST; }
  float* mixed = W + o; o += SLOT_ST;
  WSRef ws;
  ws.xg = W + o; o += SLOT_PRE;
  ws.xt = W + o; o += SLOT_T;
  ws.t1 = W + o; o += SLOT_T;
  ws.t2 = W + o; o += SLOT_T;
  ws.outAcc = W + o; o += SLOT_T;
  ws.feat = W + o; o += 16384;
  ws.gate = W + o; o += 4096;
  ws.mean = W + o; o += 64;
  ws.rstd = W + o; o += 64;
  float* wn = W + o; o += 140; float* wr = W + o; o += 140;
  float* Gn = W + o; o += 100; float* Gr = W + o; o += 100;
  float* w2n = W + o; o += 16; float* w2r = W + o; o += 16;
  ws.idx = (int*)(W + o); o += 16;

  // ---- parameter offsets (jax tree-leaves order) ----
  CellOff cells[8];
  int cellC[8]; bool cellRed[8], cellRp[8];
  {
    int cpp = 48, cp = 48, cc = 16; bool rp = false; size_t cur = 0;
    for (int i = 0; i < 8; ++i) {
      bool red = (i == 2) || (i == 5);
      if (red) cc *= 2;
      cellC[i] = cc; cellRed[i] = red; cellRp[i] = rp;
      cur = walkCell(cur, cc, red, rp, cpp, cp, cells[i]);
      rp = red; cpp = cp; cp = 4 * cc;
    }
    // cls{b,w}, stem{b,g,w}
    size_t clsB = cur; cur += 10;
    size_t clsW = cur; cur += (size_t)10 * 4 * cc;
    size_t stemB = cur; cur += 48;
    size_t stemG = cur; cur += 48;
    size_t stemW = cur;

    // ---- alpha / beta preprocessing ----
    k_softmax10<<<14, 1, 0, stream>>>(an, wn);
    k_softmax10<<<14, 1, 0, stream>>>(ar, wr);
    k_gram<<<1, 128, 0, stream>>>(wn, Gn);
    k_gram<<<1, 128, 0, stream>>>(wr, Gr);
    k_beta<<<1, 1, 0, stream>>>(bnv, w2n);
    k_beta<<<1, 1, 0, stream>>>(brv, w2r);

    // ---- stem: 3x3 conv 3->48 pad 1 + affine BN ----
    conv(x, P + stemW, slots[0], Nb, 3, 48, 32, 32, 3, 3, 1, 1, 1, 1, 1, 1, 1, 0, stream);
    bnrun(slots[0], Nb, 48, 1024, ws.mean, ws.rstd, P + stemG, P + stemB, stream);

    float* s0 = slots[0]; float* s1 = slots[0];
    int d0C = 48, d0H = 32, d0W = 32, d1C = 48, d1H = 32, d1W = 32;

    for (int L = 0; L < 8; ++L) {
      int C = cellC[L], e;
      bool red = cellRed[L], rp2 = cellRp[L];
      CellOff& CO = cells[L];
      // preprocess s0
      int p0H = d0H, p0W = d0W;
      if (rp2) {
        pw(s0, P + CO.p0a, pre0, Nb, d0C, C / 2, d0H, d0W, 2, 0, 0, C, 1, stream);
        pw(s0, P + CO.p0b, pre0, Nb, d0C, C / 2, d0H, d0W, 2, 1, C / 2, C, 1, stream);
        p0H /= 2; p0W /= 2;
      } else {
        pw(s0, P + CO.p0a, pre0, Nb, d0C, C, d0H, d0W, 1, 0, 0, C, 1, stream);
      }
      bnrun(pre0, Nb, C, p0H * p0W, ws.mean, ws.rstd, nullptr, nullptr, stream);
      // preprocess s1
      pw(s1, P + CO.p1, pre1, Nb, d1C, C, d1H, d1W, 1, 0, 0, C, 1, stream);
      bnrun(pre1, Nb, C, d1H * d1W, ws.mean, ws.rstd, nullptr, nullptr, stream);

      const float* wv = red ? wr : wn;
      const float* Gm = red ? Gr : Gn;
      const float* w2v = red ? w2r : w2n;
      int Ho = d1H / (red ? 2 : 1), Wo = d1W / (red ? 2 : 1);
      float* stB[6]; int stH[6], stW[6];
      stB[0] = pre0; stH[0] = p0H; stW[0] = p0W;
      stB[1] = pre1; stH[1] = d1H; stW[1] = d1W;
      e = 0;
      for (int i = 0; i < 4; ++i) {
        float* acc = st4[i];
        size_t nAcc = (size_t)Nb * C * Ho * Wo;
        k_zero<<<gblk(nAcc), 256, 0, stream>>>(acc, nAcc);
        for (int j = 0; j < 2 + i; ++j, ++e) {
          int stride = (red && j < 2) ? 2 : 1;
          runMixed(P, stB[j], Nb, C, stH[j], stW[j], stride, CO.e[e],
                   wv, e * 10, Gm, mixed, ws, stream);
          k_axpy<<<gblk(nAcc), 256, 0, stream>>>(acc, mixed, nAcc, w2v, e);
        }
        stB[2 + i] = acc; stH[2 + i] = Ho; stW[2 + i] = Wo;
      }
      // concat states[2..5] -> new s1
      float* ns = slots[0];
      for (int q = 0; q < 3; ++q)
        if (slots[q] != s0 && slots[q] != s1) { ns = slots[q]; break; }
      for (int q = 0; q < 4; ++q)
        k_copy_ch<<<gblk((size_t)Nb * C * Ho * Wo), 256, 0, stream>>>(
            ns, st4[q], q * C, C, 4 * C, Ho * Wo, Nb);
      s0 = s1; d0C = d1C; d0H = d1H; d0W = d1W;
      s1 = ns; d1C = 4 * C; d1H = Ho; d1W = Wo;
    }
    // ---- classifier ----
    k_head<<<Nb, 256, 0, stream>>>(s1, P + clsW, P + clsB, out, Nb, d1C, d1H * d1W);
  }
  (void)n_in; (void)out_size; (void)ws_size;
}